// GCN_12086037971340
// MI455X (gfx1250) — compile-verified
//
#include <hip/hip_runtime.h>
#include <math.h>

// ---------------------------------------------------------------------------
// GCN pipeline for MI455X (gfx1250, wave32)
//   dense GEMMs     -> V_WMMA_F32_16X16X4_F32, 16x(16*TN) register tile/wave,
//                      software-pipelined (double-buffered A/B fragments)
//   edge aggregation-> coalesced float4 gather + global_atomic_add_f32
// ---------------------------------------------------------------------------

#define NN   50000
#define EE   800000
#define IND  64
#define HH   128
#define OUTD 16

typedef float v2f __attribute__((ext_vector_type(2)));
typedef float v8f __attribute__((ext_vector_type(8)));

__device__ __forceinline__ float elu_f(float v) {
    return v > 0.0f ? v : (__expf(v) - 1.0f);
}

// ---------------------------------------------------------------------------
// GEMM: C[M x Nc] = act(A[M x K] @ W[K x Nc] + bias)
// One wave computes a 16 x (16*TN) output strip. Per k-step: 1 float2 A load
// (reused TN times) + 2*TN B loads, TN chained V_WMMA_F32_16X16X4_F32.
// The loop is software-pipelined: fragments for k-step i+1 are loaded before
// the WMMAs of k-step i issue, so WMMAs wait only on the *previous* group
// (s_wait_loadcnt > 0) instead of their own loads.
// Layouts per CDNA5 ISA 7.12.2:
//   A 16x4 : row = lane&15, k = 2*(lane>>4) + vgpr          (2 VGPRs)
//   B 4x16 : col = lane&15, k = 2*(lane>>4) + vgpr          (2 VGPRs)
//   C 16x16: col = lane&15, row = 8*(lane>>4) + vgpr        (8 VGPRs)
// M multiple of 16 (50000 = 3125*16), K multiple of 4, Nc multiple of 16*TN.
// EXEC is all-ones for every wave that reaches the WMMAs (wave-uniform exit).
// ---------------------------------------------------------------------------
template <int ACT, int TN>
__global__ void gemm16_wmma(const float* __restrict__ A,
                            const float* __restrict__ W,
                            const float* __restrict__ bias,
                            float* __restrict__ C,
                            int M, int K, int Nc) {
    int gid  = blockIdx.x * blockDim.x + threadIdx.x;
    int wave = gid >> 5;
    int lane = gid & 31;
    int groupsN = Nc / (16 * TN);
    int tilesM  = M >> 4;
    if (wave >= tilesM * groupsN) return;   // wave-uniform exit
    int tm = wave / groupsN;
    int tg = wave - tm * groupsN;

    int r  = lane & 15;          // A: row index / B,C: column index
    int kq = (lane >> 4) << 1;   // k sub-pair base: 0 or 2

    const float* Arow = A + (size_t)(tm * 16 + r) * K + kq;
    const float* Bp   = W + (size_t)kq * Nc + tg * (16 * TN) + r;
    const size_t bstep = (size_t)4 * Nc;

    v8f acc[TN];
#pragma unroll
    for (int t = 0; t < TN; ++t) acc[t] = (v8f){};

    // --- pipeline prologue: load fragments for k = 0 ---
    v2f a_cur, b_cur[TN];
    {
        float2 av = *(const float2*)(Arow);
        a_cur.x = av.x;
        a_cur.y = av.y;
#pragma unroll
        for (int t = 0; t < TN; ++t) {
            b_cur[t].x = Bp[t * 16];
            b_cur[t].y = Bp[Nc + t * 16];
        }
    }

    // --- steady state: load k0, compute k0-4 ---
    for (int k0 = 4; k0 < K; k0 += 4) {
        const float* Bn = Bp + bstep;
        float2 an = *(const float2*)(Arow + k0);
        v2f a_nxt, b_nxt[TN];
        a_nxt.x = an.x;
        a_nxt.y = an.y;
#pragma unroll
        for (int t = 0; t < TN; ++t) {
            b_nxt[t].x = Bn[t * 16];
            b_nxt[t].y = Bn[Nc + t * 16];
        }
#pragma unroll
        for (int t = 0; t < TN; ++t) {
            // 8 args: (neg_a, A, neg_b, B, c_mod, C, reuse_a, reuse_b)
            acc[t] = __builtin_amdgcn_wmma_f32_16x16x4_f32(
                false, a_cur, false, b_cur[t], (short)0, acc[t], false, false);
        }
        a_cur = a_nxt;
#pragma unroll
        for (int t = 0; t < TN; ++t) b_cur[t] = b_nxt[t];
        Bp = Bn;
    }

    // --- epilogue: compute last k group ---
#pragma unroll
    for (int t = 0; t < TN; ++t) {
        acc[t] = __builtin_amdgcn_wmma_f32_16x16x4_f32(
            false, a_cur, false, b_cur[t], (short)0, acc[t], false, false);
    }

    int mb = tm * 16 + ((lane >> 4) << 3);
#pragma unroll
    for (int t = 0; t < TN; ++t) {
        int n = tg * (16 * TN) + t * 16 + r;
        float bv = bias ? bias[n] : 0.0f;
#pragma unroll
        for (int g = 0; g < 8; ++g) {
            float v = acc[t][g] + bv;
            if (ACT) v = elu_f(v);
            C[(size_t)(mb + g) * Nc + n] = v;
        }
    }
}

// ---------------------------------------------------------------------------
// Elementwise / edge kernels
// ---------------------------------------------------------------------------
__global__ void fill_kernel(float* __restrict__ p, float val, int n) {
    int i = blockIdx.x * blockDim.x + threadIdx.x;
    if (i < n) p[i] = val;
}

__global__ void deg_accum_kernel(const int* __restrict__ dst,
                                 const float* __restrict__ w,
                                 float* __restrict__ deg) {
    int e = blockIdx.x * blockDim.x + threadIdx.x;
    if (e < EE) atomicAdd(&deg[dst[e]], w[e]);
}

// deg -> dinv (in place) and self_coef = 2*dinv*dinv
__global__ void dinv_kernel(float* __restrict__ deg, float* __restrict__ sc) {
    int n = blockIdx.x * blockDim.x + threadIdx.x;
    if (n < NN) {
        float di = rsqrtf(deg[n]);
        deg[n] = di;
        sc[n]  = 2.0f * di * di;
    }
}

__global__ void norm_kernel(const int* __restrict__ src,
                            const int* __restrict__ dst,
                            const float* __restrict__ w,
                            const float* __restrict__ dinv,
                            float* __restrict__ nrm) {
    int e = blockIdx.x * blockDim.x + threadIdx.x;
    if (e < EE) nrm[e] = w[e] * dinv[src[e]] * dinv[dst[e]];
}

// agg[dst] += t[src] * norm   (32 threads per edge, float4 per thread)
__global__ void scatter_kernel(const float* __restrict__ t,
                               const int* __restrict__ src,
                               const int* __restrict__ dst,
                               const float* __restrict__ nrm,
                               float* __restrict__ agg) {
    long long idx = (long long)blockIdx.x * blockDim.x + threadIdx.x;
    long long e = idx >> 5;               // H/4 = 32 feature groups per edge
    if (e >= EE) return;
    int fg = (int)(idx & 31);
    int s = src[e], d = dst[e];
    float w = nrm[e];
    const float4 tv = *(const float4*)(t + (size_t)s * HH + fg * 4);
    float* ap = agg + (size_t)d * HH + fg * 4;
    atomicAdd(ap + 0, tv.x * w);
    atomicAdd(ap + 1, tv.y * w);
    atomicAdd(ap + 2, tv.z * w);
    atomicAdd(ap + 3, tv.w * w);
}

// h = elu(agg + self_coef[n] * t + bias[f])   (in-place safe: h only written)
__global__ void post_kernel(const float* __restrict__ agg,
                            const float* __restrict__ t,
                            const float* __restrict__ sc,
                            const float* __restrict__ bias,
                            float* __restrict__ h) {
    int idx = blockIdx.x * blockDim.x + threadIdx.x;
    if (idx >= NN * HH) return;
    int n = idx >> 7;      // / 128
    int f = idx & 127;     // % 128
    float v = agg[idx] + sc[n] * t[idx] + bias[f];
    h[idx] = elu_f(v);
}

// ---------------------------------------------------------------------------
// Host-side launch
// ---------------------------------------------------------------------------
static inline int nblocks(long long t, int bs) { return (int)((t + bs - 1) / bs); }

extern "C" void kernel_launch(void* const* d_in, const int* in_sizes, int n_in,
                              void* d_out, int out_size, void* d_ws, size_t ws_size,
                              hipStream_t stream) {
    const float* x  = (const float*)d_in[0];
    const int*   ei = (const int*)d_in[1];
    const float* ew = (const float*)d_in[2];
    const float* W1 = (const float*)d_in[3];  const float* b1 = (const float*)d_in[4];
    const float* W2 = (const float*)d_in[5];  const float* b2 = (const float*)d_in[6];
    const float* W3 = (const float*)d_in[7];  const float* b3 = (const float*)d_in[8];
    const float* Wg[4] = { (const float*)d_in[9],  (const float*)d_in[11],
                           (const float*)d_in[13], (const float*)d_in[15] };
    const float* bg[4] = { (const float*)d_in[10], (const float*)d_in[12],
                           (const float*)d_in[14], (const float*)d_in[16] };
    const float* Wh = (const float*)d_in[17]; const float* bh = (const float*)d_in[18];
    float* out = (float*)d_out;

    const int* srcI = ei;        // edge_index[0]
    const int* dstI = ei + EE;   // edge_index[1]

    // Workspace layout (floats): h | t | agg | deg->dinv | self_coef | norm
    float* f    = (float*)d_ws;
    float* hbuf = f;
    float* tbuf = hbuf + (size_t)NN * HH;
    float* aggb = tbuf + (size_t)NN * HH;
    float* degb = aggb + (size_t)NN * HH;
    float* scb  = degb + NN;
    float* nrmb = scb + NN;

    const int BS = 256;

    // --- GCN normalization precompute (improved=True: self-loop weight 2.0) ---
    fill_kernel<<<nblocks(NN, BS), BS, 0, stream>>>(degb, 2.0f, NN);
    deg_accum_kernel<<<nblocks(EE, BS), BS, 0, stream>>>(dstI, ew, degb);
    dinv_kernel<<<nblocks(NN, BS), BS, 0, stream>>>(degb, scb);
    norm_kernel<<<nblocks(EE, BS), BS, 0, stream>>>(srcI, dstI, ew, degb, nrmb);

    // GEMM launchers: Nc=128 path uses 16x64 strips (TN=4, A reused 4x),
    // head (Nc=16) uses TN=1.
    auto gemm128 = [&](const float* A, const float* W, const float* bias, float* C,
                       int K, int act) {
        long long thr = (long long)(NN >> 4) * (HH / 64) * 32;
        if (act) gemm16_wmma<1, 4><<<nblocks(thr, BS), BS, 0, stream>>>(A, W, bias, C, NN, K, HH);
        else     gemm16_wmma<0, 4><<<nblocks(thr, BS), BS, 0, stream>>>(A, W, bias, C, NN, K, HH);
    };

    // --- embedding MLP: h = elu(elu(elu(x W1+b1) W2+b2) W3+b3) ---
    gemm128(x,    W1, b1, hbuf, IND, 1);
    gemm128(hbuf, W2, b2, tbuf, HH,  1);
    gemm128(tbuf, W3, b3, hbuf, HH,  1);

    // --- 4 GCNConv layers ---
    for (int l = 0; l < 4; ++l) {
        gemm128(hbuf, Wg[l], nullptr, tbuf, HH, 0);                            // t = h @ W
        fill_kernel<<<nblocks((long long)NN * HH, BS), BS, 0, stream>>>(aggb, 0.0f, NN * HH);
        scatter_kernel<<<nblocks((long long)EE * 32, BS), BS, 0, stream>>>(
            tbuf, srcI, dstI, nrmb, aggb);                                     // agg[dst] += t[src]*norm
        post_kernel<<<nblocks((long long)NN * HH, BS), BS, 0, stream>>>(
            aggb, tbuf, scb, bg[l], hbuf);                                     // h = elu(agg + sc*t + b)
    }

    // --- head: out = h @ Wh + bh ---
    {
        long long thr = (long long)(NN >> 4) * (OUTD >> 4) * 32;
        gemm16_wmma<0, 1><<<nblocks(thr, BS), BS, 0, stream>>>(hbuf, Wh, bh, out, NN, HH, OUTD);
    }
}